// GCN_75179107549512
// MI455X (gfx1250) — compile-verified
//
#include <hip/hip_runtime.h>

typedef __attribute__((ext_vector_type(2))) float v2f;
typedef __attribute__((ext_vector_type(8))) float v8f;

#define NCOLS 128
#define MTILE 32
#define LDSTRIDE 260   // 260 % 64 banks == 4 -> 16 rows hit distinct banks

// ---------------------------------------------------------------------------
// Pack weights: W (Kw x 128) -> Wp pair-interleaved, zero-padded to Kpad rows.
//   Wp[((k>>1)*128 + n)*2 + (k&1)] = (k < Kw) ? W[k*128 + n] : 0
// A lane's (B[kb][n], B[kb+1][n]) becomes one contiguous 8B load.
// ---------------------------------------------------------------------------
__global__ void pack_w(const float* __restrict__ W, float* __restrict__ Wp,
                       int Kw, int Kpad) {
  int i = blockIdx.x * blockDim.x + threadIdx.x;
  if (i < Kpad * NCOLS) {
    int k = i / NCOLS;
    int n = i - k * NCOLS;
    float v = (k < Kw) ? W[(size_t)k * NCOLS + n] : 0.0f;
    Wp[((size_t)(k >> 1) * NCOLS + n) * 2 + (k & 1)] = v;
  }
}

// ---------------------------------------------------------------------------
// Init: deg = 1 (self loop), s1 = 0, agg2 = 0.
// ---------------------------------------------------------------------------
__global__ void init_ws(float* __restrict__ deg, float* __restrict__ s1,
                        float* __restrict__ agg2, int N) {
  unsigned tid = blockIdx.x * blockDim.x + threadIdx.x;
  if (tid < (unsigned)N * 128u) {
    agg2[tid] = 0.0f;
    unsigned j = tid & 127u;
    unsigned i = tid >> 7;
    if (j == 0u) deg[i] = 1.0f;
    if (j < 5u)  s1[i * 5u + j] = 0.0f;
  }
}

// ---------------------------------------------------------------------------
// Layer-1 edge pass: deg[dst] += 1 ; s1[dst] += x[src]  (5 features)
// ---------------------------------------------------------------------------
__global__ void edge_accum_feat(const int* __restrict__ src, const int* __restrict__ dst,
                                const float* __restrict__ x, float* __restrict__ deg,
                                float* __restrict__ s1, int E) {
  int e = blockIdx.x * blockDim.x + threadIdx.x;
  if (e < E) {
    int s = src[e], d = dst[e];
    atomicAdd(&deg[d], 1.0f);
#pragma unroll
    for (int f = 0; f < 5; ++f)
      atomicAdd(&s1[(size_t)d * 5 + f], x[(size_t)s * 5 + f]);
  }
}

// ---------------------------------------------------------------------------
// Layer-2 edge pass: agg2[dst][0..127] += h1[src][0..127]
// One wave per edge, float4 per lane: coalesced 512B read per edge.
// ---------------------------------------------------------------------------
__global__ void edge_accum_embed(const int* __restrict__ src, const int* __restrict__ dst,
                                 const float* __restrict__ h, float* __restrict__ agg, int E) {
  unsigned tid = blockIdx.x * blockDim.x + threadIdx.x;
  unsigned e = tid >> 5;
  unsigned c = (tid & 31u) * 4u;
  if (e < (unsigned)E) {
    int s = src[e], d = dst[e];
    const float4 v = *(const float4*)(h + (size_t)s * 128 + c);
    float* a = agg + (size_t)d * 128 + c;
    atomicAdd(a + 0, v.x);
    atomicAdd(a + 1, v.y);
    atomicAdd(a + 2, v.z);
    atomicAdd(a + 3, v.w);
  }
}

// ---------------------------------------------------------------------------
// SAGE GEMM via V_WMMA_F32_16X16X4_F32, branch-free inner loop.
//   A row = [A0[row] | (A0[row]+A1[row])/deg[row]]  (cat mode, A1 != null)
//         =  A0[row]                                 (plain mode)
//   out = relu(A @ W + bias), Wp pre-packed/padded, out is M x 128.
// Block: 256 threads = 8 waves covering 32 rows x 128 cols.
// Each wave: two 16x16 row tiles sharing each B load (2 WMMAs / B load).
// Full tiles (the common case) take an unguarded, EXEC-clean store path.
// ---------------------------------------------------------------------------
__global__ void sage_gemm_wmma(const float* __restrict__ A0, const float* __restrict__ A1,
                               const float* __restrict__ deg, const float* __restrict__ Wp,
                               const float* __restrict__ bias, float* __restrict__ out,
                               int M, int d, int Kpad) {
  __shared__ float As[MTILE * LDSTRIDE];

  const int row0 = blockIdx.x * MTILE;
  const int t = threadIdx.x;

  // Stage the 32 x Kpad A-tile (zero padded rows / K) into LDS.
  const int total = MTILE * Kpad;
  for (int i = t; i < total; i += 256) {
    int m = i / Kpad;
    int k = i - m * Kpad;
    int row = row0 + m;
    float v = 0.0f;
    if (row < M) {
      if (k < d) {
        v = A0[(size_t)row * d + k];
      } else if (A1 != nullptr && k < 2 * d) {
        int kk = k - d;
        v = (A0[(size_t)row * d + kk] + A1[(size_t)row * d + kk]) / deg[row];
      }
    }
    As[m * LDSTRIDE + k] = v;
  }
  __syncthreads();

  const int lane  = t & 31;
  const int wave  = t >> 5;                 // 8 waves -> 8 column tiles of 16
  const int n     = wave * 16 + (lane & 15);
  const int mA    = lane & 15;
  const int khalf = (lane < 16) ? 0 : 2;    // K split across half-waves

  v8f c0 = {};
  v8f c1 = {};
#pragma unroll 2
  for (int k0 = 0; k0 < Kpad; k0 += 4) {
    const int kb = k0 + khalf;              // even
    v2f a0 = *(const v2f*)&As[mA * LDSTRIDE + kb];
    v2f a1 = *(const v2f*)&As[(mA + 16) * LDSTRIDE + kb];
    v2f b  = *(const v2f*)(Wp + ((size_t)(kb >> 1) * NCOLS + n) * 2);
    c0 = __builtin_amdgcn_wmma_f32_16x16x4_f32(false, a0, false, b,
                                               (short)0, c0, false, false);
    c1 = __builtin_amdgcn_wmma_f32_16x16x4_f32(false, a1, false, b,
                                               (short)0, c1, false, false);
  }

  const float bv    = bias[n];
  const int   rbase = (lane < 16) ? 0 : 8;  // C/D layout: vgpr r -> row r / r+8
  float* outA = out + (size_t)(row0 + rbase) * NCOLS + n;
  float* outB = outA + (size_t)16 * NCOLS;

  if (row0 + MTILE <= M) {
    // Full tile (uniform test): unguarded stores, no EXEC churn.
#pragma unroll
    for (int r = 0; r < 8; ++r) {
      float vA = c0[r] + bv;
      float vB = c1[r] + bv;
      outA[(size_t)r * NCOLS] = vA > 0.0f ? vA : 0.0f;
      outB[(size_t)r * NCOLS] = vB > 0.0f ? vB : 0.0f;
    }
  } else {
#pragma unroll
    for (int r = 0; r < 8; ++r) {
      int rowA = row0 + rbase + r;
      if (rowA < M) {
        float v = c0[r] + bv;
        outA[(size_t)r * NCOLS] = v > 0.0f ? v : 0.0f;
      }
      int rowB = row0 + 16 + rbase + r;
      if (rowB < M) {
        float v = c1[r] + bv;
        outB[(size_t)r * NCOLS] = v > 0.0f ? v : 0.0f;
      }
    }
  }
}

// ---------------------------------------------------------------------------
// Final head: out[i][0..1] = h3[i] @ lin2_W + lin2_b  (no relu)
// ---------------------------------------------------------------------------
__global__ void lin2_kernel(const float* __restrict__ h, const float* __restrict__ W,
                            const float* __restrict__ b, float* __restrict__ out, int M) {
  int i = blockIdx.x * blockDim.x + threadIdx.x;
  if (i < M) {
    float a0 = b[0], a1 = b[1];
    const float* row = h + (size_t)i * 128;
#pragma unroll 8
    for (int k = 0; k < 128; ++k) {
      float v = row[k];
      a0 += v * W[2 * k];
      a1 += v * W[2 * k + 1];
    }
    out[(size_t)i * 2 + 0] = a0;
    out[(size_t)i * 2 + 1] = a1;
  }
}

extern "C" void kernel_launch(void* const* d_in, const int* in_sizes, int n_in,
                              void* d_out, int out_size, void* d_ws, size_t ws_size,
                              hipStream_t stream) {
  const float* x   = (const float*)d_in[0];
  const int*   ei  = (const int*)d_in[1];
  const float* W1  = (const float*)d_in[2];
  const float* b1  = (const float*)d_in[3];
  const float* W2  = (const float*)d_in[4];
  const float* b2  = (const float*)d_in[5];
  const float* l1W = (const float*)d_in[6];
  const float* l1b = (const float*)d_in[7];
  const float* l2W = (const float*)d_in[8];
  const float* l2b = (const float*)d_in[9];

  const int N = in_sizes[0] / 5;
  const int E = in_sizes[1] / 2;
  const int* src = ei;
  const int* dst = ei + E;

  // Workspace (floats):
  //   deg(N) | s1(5N) | agg2(128N) | h1(128N) | h2(128N) | Wp1(12*128) | Wp2(256*128) | Wp3(128*128)
  float* ws   = (float*)d_ws;
  float* deg  = ws;
  float* s1   = deg  + (size_t)N;
  float* agg2 = s1   + (size_t)5 * N;
  float* h1   = agg2 + (size_t)128 * N;
  float* h2   = h1   + (size_t)128 * N;
  float* Wp1  = h2   + (size_t)128 * N;
  float* Wp2  = Wp1  + (size_t)12 * 128;
  float* Wp3  = Wp2  + (size_t)256 * 128;
  float* h3   = h1;  // h1 dead after layer-2 GEMM; reuse for lin1 output

  const int nthread = 256;

  // 0) pack / pad weights into WMMA-friendly pair-interleaved layout
  pack_w<<<(12  * NCOLS + nthread - 1) / nthread, nthread, 0, stream>>>(W1,  Wp1, 10,  12);
  pack_w<<<(256 * NCOLS + nthread - 1) / nthread, nthread, 0, stream>>>(W2,  Wp2, 256, 256);
  pack_w<<<(128 * NCOLS + nthread - 1) / nthread, nthread, 0, stream>>>(l1W, Wp3, 128, 128);

  // 1) init deg/s1/agg2
  init_ws<<<((unsigned)N * 128u + nthread - 1) / nthread, nthread, 0, stream>>>(deg, s1, agg2, N);

  // 2) layer-1 edge aggregation (degree + 5-feature sums)
  edge_accum_feat<<<(E + nthread - 1) / nthread, nthread, 0, stream>>>(src, dst, x, deg, s1, E);

  // 3) h1 = relu([x | (s1+x)/deg] @ W1 + b1)      K = 10 -> padded to 12
  sage_gemm_wmma<<<(N + MTILE - 1) / MTILE, nthread, 0, stream>>>(x, s1, deg, Wp1, b1, h1, N, 5, 12);

  // 4) layer-2 edge aggregation: agg2[dst] += h1[src]   (bandwidth-dominant step)
  edge_accum_embed<<<((unsigned)E * 32u + nthread - 1) / nthread, nthread, 0, stream>>>(src, dst, h1, agg2, E);

  // 5) h2 = relu([h1 | (agg2+h1)/deg] @ W2 + b2)  K = 256
  sage_gemm_wmma<<<(N + MTILE - 1) / MTILE, nthread, 0, stream>>>(h1, agg2, deg, Wp2, b2, h2, N, 128, 256);

  // 6) h3 = relu(h2 @ lin1_W + lin1_b)            K = 128, plain A
  sage_gemm_wmma<<<(N + MTILE - 1) / MTILE, nthread, 0, stream>>>(h2, nullptr, deg, Wp3, l1b, h3, N, 128, 128);

  // 7) out = h3 @ lin2_W + lin2_b
  lin2_kernel<<<(N + nthread - 1) / nthread, nthread, 0, stream>>>(h3, l2W, l2b, (float*)d_out, N);
}